// Spatial_Weighting_84885733638230
// MI455X (gfx1250) — compile-verified
//
#include <hip/hip_runtime.h>
#include <math.h>
#include <stdint.h>

// Problem constants (qkv: (3, 32, 12, 256, 64) f32, num_frames = 8)
#define TN 2048        // rows per sim batch = num_frames * 256
#define CD 768         // feature dim = 12 * 64
#define NB 2           // number of sim batches (b // 2)
#define T_ITERS 150    // deflated power-iteration steps replacing eigh
#define NSTAGE 96      // 2 gram sources * (768/16) k-stages

typedef __attribute__((ext_vector_type(2))) float v2f;
typedef __attribute__((ext_vector_type(8))) float v8f;

__device__ __forceinline__ v8f wmma_f32_16x16x4(v2f a, v2f b, v8f c) {
  // D = A(16x4, f32) * B(4x16, f32) + C(16x16, f32)
  return __builtin_amdgcn_wmma_f32_16x16x4_f32(false, a, false, b, (short)0, c,
                                               false, false);
}

// CDNA5 async DMA: global -> LDS without a VGPR round-trip, tracked by ASYNCcnt.
// vdst VGPR carries the per-lane LDS byte offset (low 32 bits of the generic
// pointer to a __shared__ object == addrspace(3) offset, per ISA aperture rule).
__device__ __forceinline__ void async_load_b128(uint32_t lds_off, const float* g) {
  asm volatile("global_load_async_to_lds_b128 %0, %1, off"
               :: "v"(lds_off), "v"((unsigned long long)(uintptr_t)g)
               : "memory");
}
__device__ __forceinline__ void wait_async0() {
  asm volatile("s_wait_asynccnt 0x0" ::: "memory");
}

__device__ __forceinline__ float block_sum256(float v, float* sh) {
  int t = threadIdx.x;
  sh[t] = v; __syncthreads();
  for (int s = 128; s > 0; s >>= 1) { if (t < s) sh[t] += sh[t + s]; __syncthreads(); }
  float r = sh[0]; __syncthreads(); return r;
}
__device__ __forceinline__ float block_max256(float v, float* sh) {
  int t = threadIdx.x;
  sh[t] = v; __syncthreads();
  for (int s = 128; s > 0; s >>= 1) { if (t < s) sh[t] = fmaxf(sh[t], sh[t + s]); __syncthreads(); }
  float r = sh[0]; __syncthreads(); return r;
}

// ---------------------------------------------------------------------------
// 1) Repack qkv[1] (32,12,256,64) -> Kr[4][2048][768] row-major
//    row m = f*256 + n ; col c = h*64 + d ; batch b = bt/8, f = bt%8
// ---------------------------------------------------------------------------
__global__ void repack_k(const float* __restrict__ qkv, float* __restrict__ Kr) {
  long tid = (long)blockIdx.x * 256 + threadIdx.x;   // 4*2048*192 threads
  const int C4 = CD / 4;                             // 192 float4 per row
  int c4 = (int)(tid % C4);
  long rem = tid / C4;
  int m = (int)(rem % TN);
  int b = (int)(rem / TN);                           // 0..3
  int c = c4 << 2;
  int h = c >> 6;
  int d = c & 63;
  int f = m >> 8;
  int n = m & 255;
  const float* kbase = qkv + (long)1 * 32 * 12 * 256 * 64;  // qkv[1]
  long src = (((long)(b * 8 + f) * 12 + h) * 256 + n) * 64 + d;
  float4 v = *reinterpret_cast<const float4*>(kbase + src);
  *reinterpret_cast<float4*>(Kr + ((long)b * TN + m) * CD + c) = v;
}

// ---------------------------------------------------------------------------
// 2) Gram + threshold: W[i] = (0.5*(Kr[i]Kr[i]^T + Kr[i+2]Kr[i+2]^T) > 0.2)
//    fp32 WMMA 16x16x4; double-buffered LDS fed by async-to-LDS DMA so the
//    next stage's tile transfer overlaps this stage's 8 WMMA issues.
//    LDS columns are XOR-swizzled on bits [3:2] by (row>>2) to keep 16B async
//    deposits contiguous/aligned while spreading fragment reads across banks.
// ---------------------------------------------------------------------------
__global__ void gram_w(const float* __restrict__ Kr, float* __restrict__ W) {
  __shared__ __align__(16) float As[2][64][16];
  __shared__ __align__(16) float Bs[2][64][16];
  const int i   = blockIdx.z;
  const int mb  = blockIdx.y * 64;
  const int nb  = blockIdx.x * 64;
  const int tid = threadIdx.x;
  const int wave = tid >> 5, lane = tid & 31;
  const int msub  = wave & 3;
  const int nsub0 = (wave >> 2) << 1, nsub1 = nsub0 + 1;
  const int lrow = tid >> 2;            // cooperative loader: row 0..63
  const int lkq  = (tid & 3) << 2;      // k-quad 0/4/8/12
  const int scol = lkq ^ (((lrow >> 2) & 3) << 2);   // swizzled store column
  const int r    = lane & 15;
  const int koff = (lane >> 4) << 1;    // fp32 A/B frag: half-wave holds K+2..3
  const int rowA  = msub * 16 + r;
  const int rowB0 = nsub0 * 16 + r;
  const int rowB1 = nsub1 * 16 + r;
  const int swA  = ((rowA  >> 2) & 3) << 2;
  const int swB0 = ((rowB0 >> 2) & 3) << 2;
  const int swB1 = ((rowB1 >> 2) & 3) << 2;

  uint32_t ldsA[2], ldsB[2];
  ldsA[0] = (uint32_t)(uintptr_t)&As[0][lrow][scol];
  ldsA[1] = (uint32_t)(uintptr_t)&As[1][lrow][scol];
  ldsB[0] = (uint32_t)(uintptr_t)&Bs[0][lrow][scol];
  ldsB[1] = (uint32_t)(uintptr_t)&Bs[1][lrow][scol];

  v8f c0 = {}; v8f c1 = {};

  // prologue: stage 0 (source k1[i], k0 = 0) into buffer 0
  {
    const float* M = Kr + (long)i * TN * CD;
    async_load_b128(ldsA[0], M + (long)(mb + lrow) * CD + lkq);
    async_load_b128(ldsB[0], M + (long)(nb + lrow) * CD + lkq);
  }
  for (int st = 0; st < NSTAGE; ++st) {
    const int buf = st & 1;
    wait_async0();        // this lane's deposits for stage st have landed
    __syncthreads();      // everyone's landed; everyone done reading buf^1
    if (st + 1 < NSTAGE) {
      const int st1 = st + 1;
      const float* M = Kr + (long)(i + 2 * (st1 / 48)) * TN * CD;  // k1[i] / k2[i]
      const int k0 = (st1 % 48) * 16;
      async_load_b128(ldsA[buf ^ 1], M + (long)(mb + lrow) * CD + k0 + lkq);
      async_load_b128(ldsB[buf ^ 1], M + (long)(nb + lrow) * CD + k0 + lkq);
    }
#pragma unroll
    for (int kk = 0; kk < 16; kk += 4) {
      const int ka  = (kk + koff) ^ swA;
      const int kb0 = (kk + koff) ^ swB0;
      const int kb1 = (kk + koff) ^ swB1;
      v2f a, b0, b1;
      a.x  = As[buf][rowA][ka];    a.y  = As[buf][rowA][ka + 1];
      b0.x = Bs[buf][rowB0][kb0];  b0.y = Bs[buf][rowB0][kb0 + 1];
      b1.x = Bs[buf][rowB1][kb1];  b1.y = Bs[buf][rowB1][kb1 + 1];
      c0 = wmma_f32_16x16x4(a, b0, c0);
      c1 = wmma_f32_16x16x4(a, b1, c1);
    }
  }
  // C/D layout: VGPR rr -> M=rr (lanes 0-15) / M=rr+8 (lanes 16-31); N = lane&15
  float* Wi = W + (long)i * TN * TN;
  const int mlo = mb + msub * 16 + ((lane >> 4) << 3);
  const int n0  = nb + nsub0 * 16 + r;
  const int n1  = nb + nsub1 * 16 + r;
#pragma unroll
  for (int rr = 0; rr < 8; ++rr) {
    float s0 = 0.5f * c0[rr];
    float s1 = 0.5f * c1[rr];
    Wi[(long)(mlo + rr) * TN + n0] = (s0 > 0.2f) ? 1.0f : 1e-5f;
    Wi[(long)(mlo + rr) * TN + n1] = (s1 > 0.2f) ? 1.0f : 1e-5f;
  }
}

// ---------------------------------------------------------------------------
// 3) deg[i][m] = sum_n W[i][m][n], invd = rsqrt(deg). One wave per row.
// ---------------------------------------------------------------------------
__global__ void deg_kernel(const float* __restrict__ W, float* __restrict__ deg,
                           float* __restrict__ invd) {
  const int i = blockIdx.y;
  const int wave = threadIdx.x >> 5, lane = threadIdx.x & 31;
  const int row = blockIdx.x * 8 + wave;
  const float* Wr = W + ((long)i * TN + row) * TN;
  float s = 0.f;
  for (int j = lane; j < TN; j += 32) s += Wr[j];
#pragma unroll
  for (int o = 16; o > 0; o >>= 1) s += __shfl_xor(s, o, 32);
  if (lane == 0) { deg[i * TN + row] = s; invd[i * TN + row] = rsqrtf(s); }
}

// ---------------------------------------------------------------------------
// 4) init: u1 = sqrt(deg)/||sqrt(deg)|| (exact null vector of L),
//    x0 = deterministic pseudo-random unit vector, t = invd .* x, accum = 0.
// ---------------------------------------------------------------------------
__global__ void init_kernel(const float* __restrict__ deg, const float* __restrict__ invd,
                            float* __restrict__ u1, float* __restrict__ x,
                            float* __restrict__ t, float* __restrict__ accum) {
  __shared__ float sh[256];
  const int i = blockIdx.x;
  const int tid = threadIdx.x;
  float s = 0.f;
  for (int j = 0; j < 8; ++j) s += deg[i * TN + tid + 256 * j];
  float sdeg = block_sum256(s, sh);
  float u1s = rsqrtf(sdeg);

  float loc[8]; float ss = 0.f;
  for (int j = 0; j < 8; ++j) {
    int idx = tid + 256 * j;
    float r = sinf((float)(idx + 1) * 12.9898f) * 43758.5453f;
    r = r - floorf(r);        // [0,1)
    r = 2.f * r - 1.f;        // (-1,1)
    loc[j] = r; ss += r * r;
  }
  float nrm = block_sum256(ss, sh);
  float inv = rsqrtf(nrm);
  for (int j = 0; j < 8; ++j) {
    int idx = i * TN + tid + 256 * j;
    float xv = loc[j] * inv;
    x[idx]  = xv;
    t[idx]  = xv * invd[idx];
    u1[idx] = sqrtf(deg[idx]) * u1s;
  }
  for (int j = tid; j < 2 * T_ITERS; j += 256) accum[i * 2 * T_ITERS + j] = 0.f;
}

// ---------------------------------------------------------------------------
// 5a) y = x + invd .* (W @ t) ; accumulate S1 = u1·y, S2 = y·y (per iter slot)
// ---------------------------------------------------------------------------
__global__ void matvec_kernel(const float* __restrict__ W, const float* __restrict__ t,
                              const float* __restrict__ x, const float* __restrict__ invd,
                              const float* __restrict__ u1, float* __restrict__ y,
                              float* __restrict__ accum, int iter) {
  __shared__ float tsh[TN];
  __shared__ float p1[8], p2[8];
  const int i = blockIdx.y;
  const int tid = threadIdx.x;
  for (int j = tid; j < TN; j += 256) tsh[j] = t[i * TN + j];
  __syncthreads();
  const int wave = tid >> 5, lane = tid & 31;
  const int row = blockIdx.x * 8 + wave;
  const float* Wr = W + ((long)i * TN + row) * TN;
  float acc = 0.f;
  for (int j = lane; j < TN; j += 32) acc += Wr[j] * tsh[j];
#pragma unroll
  for (int o = 16; o > 0; o >>= 1) acc += __shfl_xor(acc, o, 32);
  if (lane == 0) {
    float yv = x[i * TN + row] + invd[i * TN + row] * acc;
    y[i * TN + row] = yv;
    p1[wave] = u1[i * TN + row] * yv;
    p2[wave] = yv * yv;
  }
  __syncthreads();
  if (tid == 0) {
    float s1 = 0.f, s2 = 0.f;
    for (int w = 0; w < 8; ++w) { s1 += p1[w]; s2 += p2[w]; }
    atomicAdd(&accum[(i * T_ITERS + iter) * 2 + 0], s1);
    atomicAdd(&accum[(i * T_ITERS + iter) * 2 + 1], s2);
  }
}

// ---------------------------------------------------------------------------
// 5b) deflate + renormalize: x = (y - S1*u1)/||.|| ; ||y-S1*u1||^2 = S2 - S1^2
// ---------------------------------------------------------------------------
__global__ void update_kernel(const float* __restrict__ y, const float* __restrict__ u1,
                              const float* __restrict__ invd, const float* __restrict__ accum,
                              float* __restrict__ x, float* __restrict__ t, int iter) {
  const int i = blockIdx.y;
  const int idx = blockIdx.x * 256 + threadIdx.x;
  const float S1 = accum[(i * T_ITERS + iter) * 2 + 0];
  const float S2 = accum[(i * T_ITERS + iter) * 2 + 1];
  const float inv = rsqrtf(fmaxf(S2 - S1 * S1, 1e-30f));
  const int g = i * TN + idx;
  float xv = (y[g] - S1 * u1[g]) * inv;
  x[g] = xv;
  t[g] = xv * invd[g];
}

// ---------------------------------------------------------------------------
// 6) v = x .* invd ; bipartition by sign of mean (sign-invariant, matches ref);
//    softmax of |v| over the kept side, zeros elsewhere.
// ---------------------------------------------------------------------------
__global__ void final_kernel(const float* __restrict__ x, const float* __restrict__ invd,
                             float* __restrict__ out) {
  __shared__ float sh[256];
  const int i = blockIdx.x;
  const int tid = threadIdx.x;
  float v[8]; float s = 0.f;
  for (int j = 0; j < 8; ++j) {
    int g = i * TN + tid + 256 * j;
    v[j] = x[g] * invd[g];
    s += v[j];
  }
  float avg = block_sum256(s, sh) * (1.f / (float)TN);
  bool pos = avg > 0.f;
  float e[8]; float mx = -INFINITY;
  for (int j = 0; j < 8; ++j) {
    bool keep = (v[j] != 0.f) && ((v[j] > 0.f) == pos);
    e[j] = keep ? fabsf(v[j]) : -INFINITY;
    mx = fmaxf(mx, e[j]);
  }
  float gmx = block_max256(mx, sh);
  float ex[8]; float ssum = 0.f;
  for (int j = 0; j < 8; ++j) {
    ex[j] = (e[j] == -INFINITY) ? 0.f : __expf(e[j] - gmx);
    ssum += ex[j];
  }
  float tot = block_sum256(ssum, sh);
  float invtot = 1.f / tot;
  for (int j = 0; j < 8; ++j) out[i * TN + tid + 256 * j] = ex[j] * invtot;
}

extern "C" void kernel_launch(void* const* d_in, const int* in_sizes, int n_in,
                              void* d_out, int out_size, void* d_ws, size_t ws_size,
                              hipStream_t stream) {
  (void)in_sizes; (void)n_in; (void)out_size; (void)ws_size;
  const float* qkv = (const float*)d_in[0];   // (3,32,12,256,64) f32; num_frames fixed = 8
  float* ws = (float*)d_ws;                   // needs ~59 MB
  float* Kr    = ws;                                   // 4*2048*768
  float* W     = Kr    + (size_t)4 * TN * CD;          // 2*2048*2048
  float* deg   = W     + (size_t)NB * TN * TN;         // 2*2048
  float* invd  = deg   + (size_t)NB * TN;
  float* u1    = invd  + (size_t)NB * TN;
  float* x     = u1    + (size_t)NB * TN;
  float* t     = x     + (size_t)NB * TN;
  float* y     = t     + (size_t)NB * TN;
  float* accum = y     + (size_t)NB * TN;              // 2 * T_ITERS * 2
  float* out   = (float*)d_out;                        // (2, 2048) f32

  repack_k  <<<dim3(6144),       dim3(256), 0, stream>>>(qkv, Kr);
  gram_w    <<<dim3(32, 32, 2),  dim3(256), 0, stream>>>(Kr, W);
  deg_kernel<<<dim3(256, 2),     dim3(256), 0, stream>>>(W, deg, invd);
  init_kernel<<<dim3(2),         dim3(256), 0, stream>>>(deg, invd, u1, x, t, accum);
  for (int it = 0; it < T_ITERS; ++it) {
    matvec_kernel<<<dim3(256, 2), dim3(256), 0, stream>>>(W, t, x, invd, u1, y, accum, it);
    update_kernel<<<dim3(8, 2),   dim3(256), 0, stream>>>(y, u1, invd, accum, x, t, it);
  }
  final_kernel<<<dim3(2), dim3(256), 0, stream>>>(x, invd, out);
}